// DotGatConv_1211180777629
// MI455X (gfx1250) — compile-verified
//
#include <hip/hip_runtime.h>
#include <math.h>

typedef __attribute__((ext_vector_type(2))) float v2f;
typedef __attribute__((ext_vector_type(8))) float v8f;

#define IN_F  128
#define OUT_F 64

// ---------------------------------------------------------------------------
// Kernel 0: zero rst output region, init segment accumulators, sigmoid(alpha)
// ---------------------------------------------------------------------------
__global__ void dg_init_kernel(float* __restrict__ out,
                               float* __restrict__ seg_sum,
                               int*   __restrict__ seg_max,
                               const float* __restrict__ lr_alpha,
                               int N, int E) {
    long long i = (long long)blockIdx.x * blockDim.x + threadIdx.x;
    long long rst = (long long)N * OUT_F;
    if (i < rst) out[i] = 0.0f;               // rst accumulator = 0
    if (i < N) {
        seg_sum[i] = 0.0f;
        seg_max[i] = (int)0x80000000;         // identity for encoded float max
        float x = lr_alpha[i];
        out[rst + E + i] = 1.0f / (1.0f + __expf(-x));   // sigmoid output
    }
}

// ---------------------------------------------------------------------------
// Kernel 1: fs = feat @ W^T via V_WMMA_F32_16X16X4_F32 (fp32-exact)
//   one wave -> one 16x16 tile of fs; 32 chained K=4 WMMAs
//   M divisible by 16 (50000 = 3125*16), N=64, K=128 -> no bounds checks,
//   EXEC stays all-ones as WMMA requires.
// ---------------------------------------------------------------------------
__global__ void dg_gemm_wmma(const float* __restrict__ feat,
                             const float* __restrict__ W,
                             float* __restrict__ fs, int M) {
    __shared__ float Wlds[OUT_F * IN_F];      // 32 KB of 320 KB WGP LDS
    for (int i = threadIdx.x; i < OUT_F * IN_F; i += blockDim.x)
        Wlds[i] = W[i];
    __syncthreads();

    const int wave = threadIdx.x >> 5;        // 8 waves / 256-thread block
    const int lane = threadIdx.x & 31;
    const int tile = blockIdx.x * 8 + wave;   // tiles = (M/16) * 4
    const int ntiles = (M >> 4) * 4;
    if (tile >= ntiles) return;               // wave-uniform

    const int tileM = tile >> 2;              // 4 N-tiles per M-tile
    const int tileN = tile & 3;
    const int half  = lane >> 4;              // 0: K+0..1, 1: K+2..3
    const int l15   = lane & 15;

    // A-frag source: row m of feat, two consecutive K per lane (b64 loads)
    const float* arow = feat + (size_t)(tileM * 16 + l15) * IN_F + half * 2;
    // B-frag source: column n of W^T == row n of W, from LDS (ds_load_b64)
    const float* brow = Wlds + (tileN * 16 + l15) * IN_F + half * 2;

    v8f c = {};
#pragma unroll
    for (int k = 0; k < IN_F; k += 4) {
        v2f a = *(const v2f*)(arow + k);
        v2f b = *(const v2f*)(brow + k);
        // 8 args: (neg_a, A, neg_b, B, c_mod, C, reuse_a, reuse_b)
        c = __builtin_amdgcn_wmma_f32_16x16x4_f32(false, a, false, b,
                                                  (short)0, c, false, false);
    }

    // C/D layout: VGPR r -> row (half*8 + r), col l15 of the tile
    float* outp = fs + ((size_t)tileM * 16 + half * 8) * OUT_F
                     + tileN * 16 + l15;
#pragma unroll
    for (int r = 0; r < 8; ++r)
        outp[(size_t)r * OUT_F] = c[r];
}

// ---------------------------------------------------------------------------
// Kernel 2: per-edge score a[e] = dot(fs[src], fs[dst]); seg-max via int-coded
//   atomicMax. 16 lanes/edge: each half-wave reads a contiguous 256B row.
// ---------------------------------------------------------------------------
__device__ __forceinline__ int float_ord(int bits) {
    return bits >= 0 ? bits : (bits ^ 0x7fffffff);   // monotone involution
}

__global__ void dg_edge_score(const float* __restrict__ fs,
                              const int* __restrict__ src,
                              const int* __restrict__ dst,
                              float* __restrict__ a_ws,
                              int*   __restrict__ seg_max, int E) {
    int t = blockIdx.x * blockDim.x + threadIdx.x;
    int e = t >> 4, j = t & 15;
    if (e >= E) return;
    int s = src[e], d = dst[e];
    const float4 xs = *((const float4*)(fs + (size_t)s * OUT_F) + j);
    const float4 xd = *((const float4*)(fs + (size_t)d * OUT_F) + j);
    float p = xs.x * xd.x + xs.y * xd.y + xs.z * xd.z + xs.w * xd.w;
    // reduce across the 16-lane half (wave32: xor masks stay in-half)
    p += __shfl_xor(p, 8, 32);
    p += __shfl_xor(p, 4, 32);
    p += __shfl_xor(p, 2, 32);
    p += __shfl_xor(p, 1, 32);
    if (j == 0) {
        a_ws[e] = p;
        atomicMax(&seg_max[d], float_ord(__float_as_int(p)));
    }
}

// ---------------------------------------------------------------------------
// Kernel 3: e = exp(a - seg_max[dst]); seg_sum[dst] += e  (one thread/edge)
// ---------------------------------------------------------------------------
__global__ void dg_edge_exp(const int* __restrict__ dst,
                            float* __restrict__ a_ws,
                            const int* __restrict__ seg_max,
                            float* __restrict__ seg_sum, int E) {
    int e = blockIdx.x * blockDim.x + threadIdx.x;
    if (e >= E) return;
    int d = dst[e];
    float mx = __int_as_float(float_ord(seg_max[d]));
    float v = __expf(a_ws[e] - mx);
    a_ws[e] = v;
    atomicAdd(&seg_sum[d], v);
}

// ---------------------------------------------------------------------------
// Kernel 4: sa = e/seg_sum[dst]; rst[dst] += sa * soft_label[src]
//   16 lanes/edge, float4 per lane, global_atomic_add_f32 scatter (L2-resident)
// ---------------------------------------------------------------------------
__global__ void dg_edge_scatter(const float* __restrict__ a_ws,
                                const float* __restrict__ seg_sum,
                                const float* __restrict__ soft_label,
                                const int* __restrict__ src,
                                const int* __restrict__ dst,
                                float* __restrict__ out_rst,
                                float* __restrict__ out_sa, int E) {
    int t = blockIdx.x * blockDim.x + threadIdx.x;
    int e = t >> 4, j = t & 15;
    if (e >= E) return;
    int s = src[e], d = dst[e];
    float sa = a_ws[e] / seg_sum[d];
    if (j == 0) out_sa[e] = sa;
    const float4 sl = *((const float4*)(soft_label + (size_t)s * OUT_F) + j);
    float* o = out_rst + (size_t)d * OUT_F + j * 4;
    atomicAdd(o + 0, sa * sl.x);
    atomicAdd(o + 1, sa * sl.y);
    atomicAdd(o + 2, sa * sl.z);
    atomicAdd(o + 3, sa * sl.w);
}

// ---------------------------------------------------------------------------
extern "C" void kernel_launch(void* const* d_in, const int* in_sizes, int n_in,
                              void* d_out, int out_size, void* d_ws, size_t ws_size,
                              hipStream_t stream) {
    const float* feat       = (const float*)d_in[0];   // [N,128]
    const float* soft_label = (const float*)d_in[1];   // [N,64]
    const float* W          = (const float*)d_in[2];   // [64,128]
    const float* lr_alpha   = (const float*)d_in[3];   // [N,1]
    const int*   src        = (const int*)d_in[4];     // [E]
    const int*   dst        = (const int*)d_in[5];     // [E]

    const int N = in_sizes[3];   // 50000
    const int E = in_sizes[4];   // 800000

    float* out     = (float*)d_out;
    float* out_rst = out;                         // [N,64]
    float* out_sa  = out + (size_t)N * OUT_F;     // [E]

    // workspace layout (floats): fs[N*64] | a/exp[E] | seg_max[N] | seg_sum[N]
    float* ws      = (float*)d_ws;
    float* fs      = ws;
    float* a_ws    = ws + (size_t)N * OUT_F;
    int*   seg_max = (int*)(a_ws + E);
    float* seg_sum = (float*)(seg_max + N);

    // 0) init outputs / accumulators / sigmoid
    {
        long long n = (long long)N * OUT_F;
        int blocks = (int)((n + 255) / 256);
        dg_init_kernel<<<blocks, 256, 0, stream>>>(out, seg_sum, seg_max,
                                                   lr_alpha, N, E);
    }
    // 1) fs = feat @ W^T  (WMMA f32 16x16x4)
    {
        int tiles = (N >> 4) * 4;
        dg_gemm_wmma<<<(tiles + 7) / 8, 256, 0, stream>>>(feat, W, fs, N);
    }
    // 2) edge dot scores + segment max
    {
        long long th = (long long)E * 16;
        dg_edge_score<<<(int)((th + 255) / 256), 256, 0, stream>>>(
            fs, src, dst, a_ws, seg_max, E);
    }
    // 3) exp + segment sum
    dg_edge_exp<<<(E + 255) / 256, 256, 0, stream>>>(dst, a_ws, seg_max,
                                                     seg_sum, E);
    // 4) normalize + weighted scatter of soft labels
    {
        long long th = (long long)E * 16;
        dg_edge_scatter<<<(int)((th + 255) / 256), 256, 0, stream>>>(
            a_ws, seg_sum, soft_label, src, dst, out_rst, out_sa, E);
    }
}